// GCN_43997644981190
// MI455X (gfx1250) — compile-verified
//
#include <hip/hip_runtime.h>
#include <hip/hip_bf16.h>

#define N_NODES 50000
#define N_EDGES 1600000
#define DIM 128
#define EPS 1e-10f

typedef __attribute__((ext_vector_type(2))) float v2f;
typedef __attribute__((ext_vector_type(8))) float v8f;

// ---------------------------------------------------------------------------
// Kernel 1: zero the aggregation accumulator (in d_ws). float4 stores.
// ---------------------------------------------------------------------------
__global__ void gcn_zero_kernel(float4* __restrict__ p, int n4) {
    int i = blockIdx.x * blockDim.x + threadIdx.x;
    if (i < n4) {
        p[i] = make_float4(0.f, 0.f, 0.f, 0.f);
    }
}

// ---------------------------------------------------------------------------
// Kernel 2: edge scatter.  One wave (32 lanes) per edge; each lane moves a
// float4 slice of the 128-dim feature, scales by the edge value, and does
// native f32 global atomics into aggr[row].  Both the gathered feature table
// (25.6 MB) and the accumulator (25.6 MB) are L2-resident on MI455X (192 MB).
// ---------------------------------------------------------------------------
__global__ __launch_bounds__(256) void gcn_scatter_kernel(
    const float* __restrict__ feat,   // [N_NODES, DIM]
    const int*   __restrict__ rows,   // [N_EDGES]
    const int*   __restrict__ cols,   // [N_EDGES]
    const float* __restrict__ vals,   // [N_EDGES]
    float*       __restrict__ aggr)   // [N_NODES, DIM]
{
    long long gid = (long long)blockIdx.x * blockDim.x + threadIdx.x;
    int e    = (int)(gid >> 5);
    int lane = (int)(gid & 31);
    if (e >= N_EDGES) return;

    int   r = rows[e];
    int   c = cols[e];
    float v = vals[e];

    const float4 f = *(const float4*)(feat + (size_t)c * DIM + lane * 4);
    float* dst = aggr + (size_t)r * DIM + lane * 4;
    // unsafeAtomicAdd lowers to global_atomic_add_f32 (no CAS loop).
    unsafeAtomicAdd(dst + 0, v * f.x);
    unsafeAtomicAdd(dst + 1, v * f.y);
    unsafeAtomicAdd(dst + 2, v * f.z);
    unsafeAtomicAdd(dst + 3, v * f.w);
}

// ---------------------------------------------------------------------------
// Kernel 3: fused WMMA GEMM (aggr @ W^T + b) + ReLU + per-row layernorm.
// Block = 256 threads = 8 waves; block covers 16 output rows x 128 cols.
// Wave w computes the 16x16 tile at columns [16w, 16w+16) via
// V_WMMA_F32_16X16X4_F32 accumulating over K=128 (32 WMMAs).
// ReLU'd tile goes to LDS; half-wave shfl_xor butterfly computes mean/var.
// ---------------------------------------------------------------------------
__global__ __launch_bounds__(256) void gcn_gemm_norm_kernel(
    const float* __restrict__ aggr,    // [N_NODES, DIM]
    const float* __restrict__ W,       // [DIM_OUT, DIM_IN] row-major
    const float* __restrict__ bias,    // [DIM]
    const float* __restrict__ scale,   // [DIM]
    const float* __restrict__ offset,  // [DIM]
    float*       __restrict__ out)     // [N_NODES, DIM]
{
    __shared__ float s_a[16][DIM];   // 8 KB ReLU'd activation tile

    const int rowBase = blockIdx.x * 16;            // 50000/16 = 3125 exact
    const int wave    = threadIdx.x >> 5;           // 0..7
    const int lane    = threadIdx.x & 31;
    const int half    = lane >> 4;                  // 0 or 1
    const int l       = lane & 15;                  // 0..15
    const int colBase = wave * 16;

    // A row for this lane: M = rowBase + l.  B column: N = colBase + l, and
    // B[k][n] = W[n][k] so B's float2 is contiguous in W's row.
    const float* arow = aggr + (size_t)(rowBase + l) * DIM;
    const float* wrow = W    + (size_t)(colBase + l) * DIM;

    v8f c = {};  // 16x16 f32 accumulator (8 VGPRs)
    #pragma unroll
    for (int k = 0; k < DIM; k += 4) {
        // VGPR0 holds K=2*half, VGPR1 holds K=2*half+1 (contiguous pair).
        v2f a  = *(const v2f*)(arow + k + 2 * half);
        v2f bm = *(const v2f*)(wrow + k + 2 * half);
        c = __builtin_amdgcn_wmma_f32_16x16x4_f32(
                /*neg_a=*/false, a, /*neg_b=*/false, bm,
                /*c_mod=*/(short)0, c, /*reuse_a=*/false, /*reuse_b=*/false);
    }

    // Bias + ReLU, park tile in LDS.  D VGPR r: M = r + 8*half, N = l.
    float bv = bias[colBase + l];
    #pragma unroll
    for (int r = 0; r < 8; ++r) {
        float v = c[r] + bv;
        v = v > 0.f ? v : 0.f;
        s_a[r + 8 * half][colBase + l] = v;
    }
    __syncthreads();

    // Layernorm: 16 threads per row (half-wave), 8 elements each.
    const int nrow = threadIdx.x >> 4;   // 0..15
    const int t    = threadIdx.x & 15;   // 0..15

    float sum = 0.f, sumsq = 0.f;
    #pragma unroll
    for (int j = 0; j < 8; ++j) {
        float v = s_a[nrow][t + 16 * j];
        sum   += v;
        sumsq += v * v;
    }
    // Butterfly reduction within the 16-lane group (wave32-safe).
    #pragma unroll
    for (int m = 8; m >= 1; m >>= 1) {
        sum   += __shfl_xor(sum,   m, 16);
        sumsq += __shfl_xor(sumsq, m, 16);
    }
    const float inv_d = 1.0f / (float)DIM;
    float mean = sum * inv_d;
    float var  = sumsq * inv_d - mean * mean + EPS;
    float rs   = rsqrtf(var);

    float* orow = out + (size_t)(rowBase + nrow) * DIM;
    #pragma unroll
    for (int j = 0; j < 8; ++j) {
        int cidx = t + 16 * j;
        float v = s_a[nrow][cidx];
        orow[cidx] = (v - mean) * scale[cidx] * rs + offset[cidx];
    }
}

// ---------------------------------------------------------------------------
// Launch
// ---------------------------------------------------------------------------
extern "C" void kernel_launch(void* const* d_in, const int* in_sizes, int n_in,
                              void* d_out, int out_size, void* d_ws, size_t ws_size,
                              hipStream_t stream) {
    const float* feat_in = (const float*)d_in[0];
    const int*   rows    = (const int*)  d_in[1];
    const int*   cols    = (const int*)  d_in[2];
    const float* vals    = (const float*)d_in[3];
    const float* W       = (const float*)d_in[4];
    const float* b       = (const float*)d_in[5];
    const float* scale   = (const float*)d_in[6];
    const float* offset  = (const float*)d_in[7];
    float*       out     = (float*)d_out;

    float* aggr = (float*)d_ws;  // N_NODES * DIM floats = 25.6 MB scratch

    // 1) zero accumulator
    {
        int n4 = N_NODES * DIM / 4;
        int blocks = (n4 + 255) / 256;
        gcn_zero_kernel<<<blocks, 256, 0, stream>>>((float4*)aggr, n4);
    }
    // 2) edge scatter: one wave per edge
    {
        long long total = (long long)N_EDGES * 32;
        int blocks = (int)((total + 255) / 256);
        gcn_scatter_kernel<<<blocks, 256, 0, stream>>>(feat_in, rows, cols, vals, aggr);
    }
    // 3) fused WMMA GEMM + bias + ReLU + layernorm
    {
        int blocks = N_NODES / 16;  // 3125
        gcn_gemm_norm_kernel<<<blocks, 256, 0, stream>>>(aggr, W, b, scale, offset, out);
    }
}